// Decoder_78254304133650
// MI455X (gfx1250) — compile-verified
//
#include <hip/hip_runtime.h>
#include <hip/hip_bf16.h>

// ---------------------------------------------------------------------------
// Attention-LSTM decoder for MI455X (gfx1250, wave32).
//
// Roofline: scan is HBM-bound (he + h_batch streamed per step: ~270MB * 256
// ~= 69 GB ~= 3 ms @ 23.3 TB/s). All GEMMs use V_WMMA_F32_16X16X4_F32 (full
// f32 precision; ~420 GFLOP total is far below the memory floor, so f32
// matrix ops are free and match the f32 reference).
//
// Round-1 fix: scalarize tile indices with readfirstlane and template the
// subtile count so no WMMA ever executes under a modified EXEC mask
// (ISA 7.12: EXEC must be all-ones for WMMA; WMMA is not skipped at EXEC==0).
// ---------------------------------------------------------------------------

#define Bn   32
#define Tn   1024
#define Ln   256
#define Hn   512
#define H2n  1024
#define H4n  2048
#define Vn   10000
#define NFv  10
#define KSv  100
#define PDv  50

typedef __attribute__((ext_vector_type(2))) float v2f;
typedef __attribute__((ext_vector_type(8))) float v8f;

__device__ __forceinline__ v8f wmma_f32_16x16x4(v2f a, v2f b, v8f c) {
  // 8 args: (neg_a, A, neg_b, B, c_mod, C, reuse_a, reuse_b)
  return __builtin_amdgcn_wmma_f32_16x16x4_f32(false, a, false, b, (short)0, c,
                                               false, false);
}

// ---------------------------------------------------------------------------
// Generic WMMA f32 GEMM:
//   D = act( A@W + (A2@W2)? + bias? + bias2? + gatherAdd? )
// Each wave computes a 16 x (16*NSUB) tile. Requires K,K2 % 4 == 0,
// M % 16 == 0, covered columns multiple of 16. Tile indices are forced into
// SGPRs via readfirstlane so every branch is scalar and EXEC is all-ones
// around every WMMA.
// VGPR layouts per CDNA5 ISA 7.12.2:
//   A 16x4 : lane L -> m = L%16, k = (L/16)*2 + {0,1}   (float2, contiguous)
//   B 4x16 : lane L -> n = L%16, k = (L/16)*2 + {0,1}
//   C 16x16: lane L, vgpr r -> m = (L/16)*8 + r, n = L%16
// ---------------------------------------------------------------------------
template <int NSUB>
__global__ __launch_bounds__(256) void gemm_wmma_f32(
    const float* __restrict__ A, int lda, const float* __restrict__ W, int ldw, int K,
    const float* __restrict__ A2, int lda2, const float* __restrict__ W2, int ldw2, int K2,
    const float* __restrict__ bias, const float* __restrict__ bias2,
    const float* __restrict__ addMat, const int* __restrict__ addIdx,
    int addIdxStride, int addLd,
    float* __restrict__ D, long long lddRow,
    int M, int nBase, int tilesN)
{
  const int lane = threadIdx.x & 31;
  // wave-uniform scalars (force SGPR so control flow is scalar)
  const int gw    = __builtin_amdgcn_readfirstlane(
                        blockIdx.x * (blockDim.x >> 5) + (threadIdx.x >> 5));
  const int tileM = gw / tilesN;
  const int tileN = gw - tileM * tilesN;
  if (tileM * 16 >= M) return;              // scalar branch, whole wave

  const int mA = tileM * 16 + (lane & 15);  // A-frag row
  const int kb = (lane >> 4) * 2;           // A/B-frag k base (0 or 2)
  const int nL = (lane & 15);               // B/C-frag column
  const int n0 = nBase + tileN * (16 * NSUB);

  v8f acc[NSUB];
  v8f zz = {};
#pragma unroll
  for (int s = 0; s < NSUB; ++s) acc[s] = zz;

  const float* Arow = A + (long long)mA * lda;
  const float* Wcol[NSUB];
#pragma unroll
  for (int s = 0; s < NSUB; ++s) Wcol[s] = W + (n0 + s * 16 + nL);

#pragma unroll 4
  for (int k = 0; k < K; k += 4) {
    v2f a = *(const v2f*)(Arow + k + kb);
#pragma unroll
    for (int s = 0; s < NSUB; ++s) {
      v2f b;
      b.x = Wcol[s][(long long)(k + kb)     * ldw];
      b.y = Wcol[s][(long long)(k + kb + 1) * ldw];
      acc[s] = wmma_f32_16x16x4(a, b, acc[s]);
    }
  }
  if (A2) {                                  // fused second GEMM pair (scalar)
    const float* A2row = A2 + (long long)mA * lda2;
    const float* W2col[NSUB];
#pragma unroll
    for (int s = 0; s < NSUB; ++s) W2col[s] = W2 + (n0 + s * 16 + nL);
#pragma unroll 4
    for (int k = 0; k < K2; k += 4) {
      v2f a = *(const v2f*)(A2row + k + kb);
#pragma unroll
      for (int s = 0; s < NSUB; ++s) {
        v2f b;
        b.x = W2col[s][(long long)(k + kb)     * ldw2];
        b.y = W2col[s][(long long)(k + kb + 1) * ldw2];
        acc[s] = wmma_f32_16x16x4(a, b, acc[s]);
      }
    }
  }

  const int mBase = tileM * 16 + (lane >> 4) * 8;
#pragma unroll
  for (int s = 0; s < NSUB; ++s) {
    int nc = n0 + s * 16 + nL;
    float bv = (bias ? bias[nc] : 0.f) + (bias2 ? bias2[nc] : 0.f);
#pragma unroll
    for (int r = 0; r < 8; ++r) {
      int m = mBase + r;
      float v = acc[s][r] + bv;
      if (addMat) {
        int row = addIdx ? addIdx[(long long)m * addIdxStride] : m;
        v += addMat[(long long)row * addLd + nc];
      }
      v = (0.f != 0.f) ? v : v;             // no-op
      D[(long long)m * lddRow + nc] = v;
    }
  }
}

// tanh-epilogue variant (kept separate so the hot he/W_yy path has no branch)
template <int NSUB>
__global__ __launch_bounds__(256) void gemm_wmma_f32_tanh(
    const float* __restrict__ A, int lda, const float* __restrict__ W, int ldw, int K,
    const float* __restrict__ A2, int lda2, const float* __restrict__ W2, int ldw2, int K2,
    const float* __restrict__ bias, const float* __restrict__ bias2,
    float* __restrict__ D, long long lddRow, int M, int nBase, int tilesN)
{
  const int lane = threadIdx.x & 31;
  const int gw    = __builtin_amdgcn_readfirstlane(
                        blockIdx.x * (blockDim.x >> 5) + (threadIdx.x >> 5));
  const int tileM = gw / tilesN;
  const int tileN = gw - tileM * tilesN;
  if (tileM * 16 >= M) return;

  const int mA = tileM * 16 + (lane & 15);
  const int kb = (lane >> 4) * 2;
  const int nL = (lane & 15);
  const int n0 = nBase + tileN * (16 * NSUB);

  v8f acc[NSUB];
  v8f zz = {};
#pragma unroll
  for (int s = 0; s < NSUB; ++s) acc[s] = zz;

  const float* Arow = A + (long long)mA * lda;
#pragma unroll 4
  for (int k = 0; k < K; k += 4) {
    v2f a = *(const v2f*)(Arow + k + kb);
#pragma unroll
    for (int s = 0; s < NSUB; ++s) {
      int n = n0 + s * 16 + nL;
      v2f b;
      b.x = W[(long long)(k + kb)     * ldw + n];
      b.y = W[(long long)(k + kb + 1) * ldw + n];
      acc[s] = wmma_f32_16x16x4(a, b, acc[s]);
    }
  }
  if (A2) {
    const float* A2row = A2 + (long long)mA * lda2;
#pragma unroll 4
    for (int k = 0; k < K2; k += 4) {
      v2f a = *(const v2f*)(A2row + k + kb);
#pragma unroll
      for (int s = 0; s < NSUB; ++s) {
        int n = n0 + s * 16 + nL;
        v2f b;
        b.x = W2[(long long)(k + kb)     * ldw2 + n];
        b.y = W2[(long long)(k + kb + 1) * ldw2 + n];
        acc[s] = wmma_f32_16x16x4(a, b, acc[s]);
      }
    }
  }

  const int mBase = tileM * 16 + (lane >> 4) * 8;
#pragma unroll
  for (int s = 0; s < NSUB; ++s) {
    int nc = n0 + s * 16 + nL;
    float bv = (bias ? bias[nc] : 0.f) + (bias2 ? bias2[nc] : 0.f);
#pragma unroll
    for (int r = 0; r < 8; ++r) {
      int m = mBase + r;
      D[(long long)m * lddRow + nc] = tanhf(acc[s][r] + bv);
    }
  }
}

// ---------------------------------------------------------------------------
// conv: conved[b,t,nf] = sum_k alpha[b, t-50+k] * conv_w[nf,k]   (XLA corr.)
// ---------------------------------------------------------------------------
__global__ __launch_bounds__(256) void conv_kernel(
    const float* __restrict__ alpha, const float* __restrict__ conv_w,
    float* __restrict__ conved)
{
  const int t0 = blockIdx.x * 256;
  const int b  = blockIdx.y;
  __shared__ float sh[256 + KSv - 1];
  for (int i = threadIdx.x; i < 256 + KSv - 1; i += 256) {
    int src = t0 - PDv + i;
    sh[i] = (src >= 0 && src < Tn) ? alpha[b * Tn + src] : 0.f;
  }
  __syncthreads();
  const int t = t0 + threadIdx.x;
  float* out = conved + ((long long)(b * Tn + t)) * NFv;
#pragma unroll
  for (int nf = 0; nf < NFv; ++nf) {
    float acc = 0.f;
    const float* wf = conv_w + nf * KSv;
    for (int k = 0; k < KSv; ++k) acc += sh[threadIdx.x + k] * wf[k];
    out[nf] = acc;
  }
}

// ---------------------------------------------------------------------------
// energy: e[b,t] = b_ee + sum_j tanh(se[b,j] + he[b,t,j] + fe[b,t,j]) * W_ee[j]
// fe[b,t,j] = b_fe[j] + sum_nf conved[b,t,nf] * W_fe[nf,j]  (fused, K=10)
// One wave per (b,t); he streamed exactly once per step (the BW floor).
// ---------------------------------------------------------------------------
__global__ __launch_bounds__(256) void energy_kernel(
    const float* __restrict__ he, const float* __restrict__ conved,
    const float* __restrict__ se, const float* __restrict__ W_fe,
    const float* __restrict__ b_fe, const float* __restrict__ W_ee,
    const float* __restrict__ b_ee, float* __restrict__ e_out)
{
  const int lane = threadIdx.x & 31;
  const int gw = blockIdx.x * (blockDim.x >> 5) + (threadIdx.x >> 5);
  const int b = gw >> 10;
  const int t = gw & (Tn - 1);

  float cv[NFv];
  const float* cvp = conved + ((long long)(b * Tn + t)) * NFv;
#pragma unroll
  for (int nf = 0; nf < NFv; ++nf) cv[nf] = cvp[nf];

  const float* hrow = he + ((long long)(b * Tn + t)) * H2n;
  const float* srow = se + (long long)b * H2n;

  float acc = 0.f;
  for (int j = lane; j < H2n; j += 32) {
    float fe = b_fe[j];
#pragma unroll
    for (int nf = 0; nf < NFv; ++nf) fe += cv[nf] * W_fe[nf * H2n + j];
    float x = srow[j] + hrow[j] + fe;
    acc += tanhf(x) * W_ee[j];
  }
#pragma unroll
  for (int off = 16; off > 0; off >>= 1) acc += __shfl_down(acc, off, 32);
  if (lane == 0) e_out[b * Tn + t] = acc + b_ee[0];
}

// ---------------------------------------------------------------------------
// masked softmax over t (max is unmasked, per reference)
// mask[b,t] = 0 iff (b < seq_lens[b]) && (t >= seq_lens[b])
// ---------------------------------------------------------------------------
__global__ __launch_bounds__(256) void softmax_kernel(
    const float* __restrict__ e, const int* __restrict__ seq_lens,
    float* __restrict__ alpha)
{
  const int b = blockIdx.x;
  const int tid = threadIdx.x;
  const int sl = seq_lens[b];
  __shared__ float red[256];

  float ev[4];
  float mx = -3.402823466e38f;
#pragma unroll
  for (int i = 0; i < 4; ++i) {
    ev[i] = e[b * Tn + tid + i * 256];
    mx = fmaxf(mx, ev[i]);
  }
  red[tid] = mx; __syncthreads();
  for (int o = 128; o > 0; o >>= 1) {
    if (tid < o) red[tid] = fmaxf(red[tid], red[tid + o]);
    __syncthreads();
  }
  mx = red[0]; __syncthreads();

  float av[4];
  float sm = 0.f;
#pragma unroll
  for (int i = 0; i < 4; ++i) {
    int t = tid + i * 256;
    float m = ((b < sl) && (t >= sl)) ? 0.f : 1.f;
    float x = __expf(ev[i] - mx) * m;
    av[i] = x; sm += x;
  }
  red[tid] = sm; __syncthreads();
  for (int o = 128; o > 0; o >>= 1) {
    if (tid < o) red[tid] += red[tid + o];
    __syncthreads();
  }
  float inv = 1.f / red[0];
#pragma unroll
  for (int i = 0; i < 4; ++i) alpha[b * Tn + tid + i * 256] = av[i] * inv;
}

// ---------------------------------------------------------------------------
// context: g[b,j] = sum_t alpha[b,t] * h_batch[b,t,j]  (coalesced over j)
// ---------------------------------------------------------------------------
__global__ __launch_bounds__(256) void context_kernel(
    const float* __restrict__ alpha, const float* __restrict__ h_batch,
    float* __restrict__ g)
{
  const int idx = blockIdx.x * 256 + threadIdx.x;   // B*H2n threads
  const int b = idx >> 10;
  const int j = idx & (H2n - 1);
  const float* hb = h_batch + ((long long)b * Tn) * H2n + j;
  const float* ar = alpha + b * Tn;
  float acc = 0.f;
#pragma unroll 4
  for (int t = 0; t < Tn; ++t) acc += ar[t] * hb[(long long)t * H2n];
  g[idx] = acc;
}

// ---------------------------------------------------------------------------
// LSTM cell (half-tanh sigmoids per reference)
// ---------------------------------------------------------------------------
__global__ __launch_bounds__(256) void lstm_kernel(
    const float* __restrict__ rec, float* __restrict__ s, float* __restrict__ c)
{
  const int idx = blockIdx.x * 256 + threadIdx.x;   // B*Hn threads
  const int b = idx >> 9;
  const int h = idx & (Hn - 1);
  const float* r = rec + (long long)b * H4n;
  float i_ = tanhf(r[h]            * 0.5f) * 0.5f + 0.5f;
  float f_ = tanhf(r[Hn + h]       * 0.5f) * 0.5f + 0.5f;
  float g_ = tanhf(r[2 * Hn + h]);
  float o_ = tanhf(r[3 * Hn + h]   * 0.5f) * 0.5f + 0.5f;
  float cn = f_ * c[idx] + i_ * g_;
  c[idx] = cn;
  s[idx] = o_ * tanhf(cn);
}

__global__ __launch_bounds__(256) void init_zero_kernel(
    float* __restrict__ s, float* __restrict__ c, float* __restrict__ alpha)
{
  const int idx = blockIdx.x * 256 + threadIdx.x;
  if (idx < Bn * Hn) { s[idx] = 0.f; c[idx] = 0.f; }
  if (idx < Bn * Tn) alpha[idx] = 0.f;
}

// ---------------------------------------------------------------------------
// host-side GEMM dispatch: NSUB=4 full tiles + (optional) 16-wide remainder
// ---------------------------------------------------------------------------
static inline void launch_gemm(hipStream_t st,
    const float* A, int lda, const float* W, int ldw, int K,
    const float* A2, int lda2, const float* W2, int ldw2, int K2,
    const float* bias, const float* bias2,
    const float* addMat, const int* addIdx, int addIdxStride, int addLd,
    float* D, long long ldd, int M, int N)
{
  int tilesM = (M + 15) / 16;
  int tilesN = N / 64;
  if (tilesN > 0) {
    int blocks = (tilesM * tilesN + 7) / 8;
    gemm_wmma_f32<4><<<blocks, 256, 0, st>>>(A, lda, W, ldw, K,
        A2, lda2, W2, ldw2, K2, bias, bias2, addMat, addIdx, addIdxStride,
        addLd, D, ldd, M, 0, tilesN);
  }
  int rem = (N - tilesN * 64) / 16;      // 0..3 subtiles of 16
  if (rem) {
    int blocks = (tilesM + 7) / 8;
    int nBase = tilesN * 64;
    if (rem == 1)
      gemm_wmma_f32<1><<<blocks, 256, 0, st>>>(A, lda, W, ldw, K,
          A2, lda2, W2, ldw2, K2, bias, bias2, addMat, addIdx, addIdxStride,
          addLd, D, ldd, M, nBase, 1);
    else if (rem == 2)
      gemm_wmma_f32<2><<<blocks, 256, 0, st>>>(A, lda, W, ldw, K,
          A2, lda2, W2, ldw2, K2, bias, bias2, addMat, addIdx, addIdxStride,
          addLd, D, ldd, M, nBase, 1);
    else
      gemm_wmma_f32<3><<<blocks, 256, 0, st>>>(A, lda, W, ldw, K,
          A2, lda2, W2, ldw2, K2, bias, bias2, addMat, addIdx, addIdxStride,
          addLd, D, ldd, M, nBase, 1);
  }
}

static inline void launch_gemm_tanh(hipStream_t st,
    const float* A, int lda, const float* W, int ldw, int K,
    const float* A2, int lda2, const float* W2, int ldw2, int K2,
    const float* bias, const float* bias2,
    float* D, long long ldd, int M, int N)
{
  int tilesM = (M + 15) / 16;
  int tilesN = N / 64;                    // all tanh uses have N % 64 == 0
  int blocks = (tilesM * tilesN + 7) / 8;
  gemm_wmma_f32_tanh<4><<<blocks, 256, 0, st>>>(A, lda, W, ldw, K,
      A2, lda2, W2, ldw2, K2, bias, bias2, D, ldd, M, 0, tilesN);
}

extern "C" void kernel_launch(void* const* d_in, const int* in_sizes, int n_in,
                              void* d_out, int out_size, void* d_ws, size_t ws_size,
                              hipStream_t stream) {
  const float* h_batch = (const float*)d_in[0];
  const int*   seq_lens= (const int*)  d_in[1];
  const int*   labels  = (const int*)  d_in[2];
  const float* W_se    = (const float*)d_in[3];
  const float* b_se    = (const float*)d_in[4];
  const float* W_he    = (const float*)d_in[5];
  const float* b_he    = (const float*)d_in[6];
  const float* W_fe    = (const float*)d_in[7];
  const float* b_fe    = (const float*)d_in[8];
  const float* W_ee    = (const float*)d_in[9];
  const float* b_ee    = (const float*)d_in[10];
  const float* conv_w  = (const float*)d_in[11];
  const float* W_sy    = (const float*)d_in[12];
  const float* b_sy    = (const float*)d_in[13];
  const float* W_gy    = (const float*)d_in[14];
  const float* b_gy    = (const float*)d_in[15];
  const float* W_yy    = (const float*)d_in[16];
  const float* b_yy    = (const float*)d_in[17];
  const float* E_yr    = (const float*)d_in[18];
  const float* W_sr    = (const float*)d_in[19];
  const float* b_sr    = (const float*)d_in[20];
  const float* W_gr    = (const float*)d_in[21];
  const float* b_gr    = (const float*)d_in[22];
  float* out = (float*)d_out;                       // (B, L, V)

  // workspace carve-out (~136.2 MB)
  char* w = (char*)d_ws;
  auto alloc = [&](size_t bytes) { void* p = w; w += (bytes + 255) & ~(size_t)255; return p; };
  float* he     = (float*)alloc((size_t)Bn * Tn * H2n * 4);  // 134 MB
  float* conved = (float*)alloc((size_t)Bn * Tn * NFv * 4);
  float* alpha  = (float*)alloc((size_t)Bn * Tn * 4);
  float* e_buf  = (float*)alloc((size_t)Bn * Tn * 4);
  float* se     = (float*)alloc((size_t)Bn * H2n * 4);
  float* g      = (float*)alloc((size_t)Bn * H2n * 4);
  float* s      = (float*)alloc((size_t)Bn * Hn * 4);
  float* c      = (float*)alloc((size_t)Bn * Hn * 4);
  float* z      = (float*)alloc((size_t)Bn * Hn * 4);
  float* rec    = (float*)alloc((size_t)Bn * H4n * 4);

  // state init
  init_zero_kernel<<<(Bn * Tn + 255) / 256, 256, 0, stream>>>(s, c, alpha);

  // he = h_batch @ W_he + b_he   (32768 x 1024 x 1024, one-time)
  launch_gemm(stream, h_batch, H2n, W_he, H2n, H2n,
              nullptr, 0, nullptr, 0, 0, b_he, nullptr,
              nullptr, nullptr, 0, 0, he, H2n, Bn * Tn, H2n);

  for (int t = 0; t < Ln; ++t) {
    // 1) conved = conv1d(alpha)
    conv_kernel<<<dim3(Tn / 256, Bn), 256, 0, stream>>>(alpha, conv_w, conved);

    // 2) se = s @ W_se + b_se          (32 x 1024 x 512)
    launch_gemm(stream, s, Hn, W_se, H2n, Hn,
                nullptr, 0, nullptr, 0, 0, b_se, nullptr,
                nullptr, nullptr, 0, 0, se, H2n, Bn, H2n);

    // 3) e = tanh(se + he + fe) @ W_ee + b_ee  (fe fused; streams he once)
    energy_kernel<<<(Bn * Tn) / 8, 256, 0, stream>>>(he, conved, se, W_fe,
                                                     b_fe, W_ee, b_ee, e_buf);

    // 4) masked softmax over t -> alpha
    softmax_kernel<<<Bn, 256, 0, stream>>>(e_buf, seq_lens, alpha);

    // 5) g = sum_t alpha * h_batch
    context_kernel<<<(Bn * H2n) / 256, 256, 0, stream>>>(alpha, h_batch, g);

    // 6) z = tanh(g@W_gy + s@W_sy + b_gy + b_sy)   (32 x 512)
    launch_gemm_tanh(stream, g, H2n, W_gy, Hn, H2n,
                     s, Hn, W_sy, Hn, Hn, b_gy, b_sy, z, Hn, Bn, Hn);

    // 7) y = z @ W_yy + b_yy  -> out[:, t, :]      (32 x 10000 x 512)
    launch_gemm(stream, z, Hn, W_yy, Vn, Hn,
                nullptr, 0, nullptr, 0, 0, b_yy, nullptr,
                nullptr, nullptr, 0, 0,
                out + (long long)t * Vn, (long long)Ln * Vn, Bn, Vn);

    // 8) rec = E_yr[labels[:,t]] + s@W_sr + g@W_gr + b_sr + b_gr  (32 x 2048)
    launch_gemm(stream, s, Hn, W_sr, H4n, Hn,
                g, H2n, W_gr, H4n, H2n, b_sr, b_gr,
                E_yr, labels + t, Ln, H4n, rec, H4n, Bn, H4n);

    // 9) LSTM cell -> s, c
    lstm_kernel<<<(Bn * Hn) / 256, 256, 0, stream>>>(rec, s, c);
  }
}